// Linear_HashEmbedder_61022895342218
// MI455X (gfx1250) — compile-verified
//
#include <hip/hip_runtime.h>

// 1-D multires hash-grid encode (Instant-NGP style), 8 levels x 2 feats.
// Key facts: resolutions are exactly 16<<lvl; max vertex index 2048 << 2^19,
// so the hash mask is identity and only the first (res+1) entries of each
// level's table are ever read: 4088 float2 entries ~= 32 KB total.
// Strategy: stage those 32 KB into LDS via gfx1250 async global->LDS copies
// (ASYNCcnt / s_wait_asynccnt), then each thread does 16 ds_load_b64 gathers
// + lerps and streams 64 B of output with non-temporal b128 stores.
// Pure bandwidth op (~272 MB HBM/call -> ~12 us at 23.3 TB/s); WMMA does not
// apply (no matmul), so the CDNA5-specific path is the async LDS data mover.

#define N_LEVELS 8
#define TABLE_ENTRIES (1 << 19)
#define THREADS 256

typedef float v2f __attribute__((ext_vector_type(2)));
typedef float v4f __attribute__((ext_vector_type(4)));
typedef int   v4i __attribute__((ext_vector_type(4)));

namespace {
// entries used per level = res + 1 (verts 0..res); padded to even count so
// every async copy is a full 16-byte b128 chunk. Reading the extra entry from
// the (huge) global table is always in-bounds.
constexpr int kPadEntries[N_LEVELS] = {18, 34, 66, 130, 258, 514, 1026, 2050};
constexpr int kOff[N_LEVELS]        = {0, 18, 52, 118, 248, 506, 1020, 2046};
constexpr int kTotalEntries = 4096; // float2 -> exactly 32 KB of LDS
} // namespace

__device__ __forceinline__ void async_copy_b128(const float* gsrc, v2f* ldst) {
#if __has_builtin(__builtin_amdgcn_global_load_async_to_lds_b128)
  // two-step casts: generic -> AS void* (addrspacecast), then AS bitcast to
  // the builtin's expected pointee type (v4i, per the compiler diagnostic).
  __attribute__((address_space(1))) void* gv =
      (__attribute__((address_space(1))) void*)(void*)const_cast<float*>(gsrc);
  __attribute__((address_space(3))) void* lv =
      (__attribute__((address_space(3))) void*)(void*)ldst;
  __builtin_amdgcn_global_load_async_to_lds_b128(
      (__attribute__((address_space(1))) v4i*)gv,
      (__attribute__((address_space(3))) v4i*)lv,
      /*offset=*/0, /*cpol=*/0);
#else
  unsigned lds_off = (unsigned)(unsigned long long)
      (__attribute__((address_space(3))) void*)(void*)ldst;
  asm volatile("global_load_async_to_lds_b128 %0, %1, off"
               :
               : "v"(lds_off), "v"(gsrc)
               : "memory");
#endif
}

__device__ __forceinline__ void wait_async_zero() {
#if __has_builtin(__builtin_amdgcn_s_wait_asynccnt)
  __builtin_amdgcn_s_wait_asynccnt(0);
#else
  asm volatile("s_wait_asynccnt 0" ::: "memory");
#endif
}

__global__ __launch_bounds__(THREADS)
void Linear_HashEmbedder_61022895342218_kernel(const float* __restrict__ x,
                                               const float* __restrict__ tables,
                                               float* __restrict__ out,
                                               int batch) {
  __shared__ v2f tab[kTotalEntries];
  const int tid = threadIdx.x;

  // ---- Stage the live table slices (32 KB) into LDS with async copies ----
#pragma unroll
  for (int lvl = 0; lvl < N_LEVELS; ++lvl) {
    const float* src = tables + (size_t)lvl * (size_t)(TABLE_ENTRIES * 2);
    const int chunks = kPadEntries[lvl] >> 1; // 16-byte chunks (2 x float2)
    for (int j = tid; j < chunks; j += THREADS) {
      async_copy_b128(src + 4 * j, &tab[kOff[lvl] + 2 * j]);
    }
  }
  wait_async_zero();
  __syncthreads();

  // ---- Grid-stride over points: 16 LDS gathers + lerps, 64 B NT store ----
  const int stride = gridDim.x * THREADS;
  for (int b = blockIdx.x * THREADS + tid; b < batch; b += stride) {
    const float xv = __builtin_nontemporal_load(x + b);
    v4f o[4];
    float* of = (float*)o;
#pragma unroll
    for (int lvl = 0; lvl < N_LEVELS; ++lvl) {
      const float res  = (float)(16 << lvl);
      const float xi   = xv * res;            // exact: res is a power of 2
      const float fidx = __builtin_floorf(xi);
      const int   idx  = (int)fidx;
      const float w    = xi - fidx;           // exact mantissa tail == ref w
      const v2f e0 = tab[kOff[lvl] + idx];
      const v2f e1 = tab[kOff[lvl] + idx + 1];
      of[2 * lvl]     = e0.x * (1.0f - w) + e1.x * w;
      of[2 * lvl + 1] = e0.y * (1.0f - w) + e1.y * w;
    }
    float* op = out + (size_t)b * 16;
    __builtin_nontemporal_store(o[0], (v4f*)(op + 0));
    __builtin_nontemporal_store(o[1], (v4f*)(op + 4));
    __builtin_nontemporal_store(o[2], (v4f*)(op + 8));
    __builtin_nontemporal_store(o[3], (v4f*)(op + 12));
  }
}

extern "C" void kernel_launch(void* const* d_in, const int* in_sizes, int n_in,
                              void* d_out, int out_size, void* d_ws, size_t ws_size,
                              hipStream_t stream) {
  (void)n_in; (void)out_size; (void)d_ws; (void)ws_size;
  const float* x      = (const float*)d_in[0];
  const float* tables = (const float*)d_in[1];
  float* out          = (float*)d_out;
  const int batch     = in_sizes[0]; // x is [B, 1]

  int blocks = (batch + THREADS - 1) / THREADS;
  if (blocks > 2048) blocks = 2048; // grid-stride; amortizes the LDS staging
  Linear_HashEmbedder_61022895342218_kernel<<<blocks, THREADS, 0, stream>>>(
      x, tables, out, batch);
}